// RelNNTransformConnected_61486751809982
// MI455X (gfx1250) — compile-verified
//
#include <hip/hip_runtime.h>
#include <hip/hip_bf16.h>

typedef __attribute__((ext_vector_type(2))) float v2f;
typedef __attribute__((ext_vector_type(8))) float v8f;

// ---------------------------------------------------------------------------
// Hardware tanh (CDNA5 TRANS op v_tanh_f32). Prefer the clang builtin when
// declared; otherwise emit the instruction directly.
// ---------------------------------------------------------------------------
__device__ __forceinline__ float fast_tanh(float x) {
#if __has_builtin(__builtin_amdgcn_tanhf)
  return __builtin_amdgcn_tanhf(x);
#elif __has_builtin(__builtin_amdgcn_tanh_f32)
  return __builtin_amdgcn_tanh_f32(x);
#else
  float r;
  asm("v_tanh_f32 %0, %1" : "=v"(r) : "v"(x));
  return r;
#endif
}

__device__ __forceinline__ v8f wmma_f32_k4(v2f a, v2f b, v8f c) {
  // D = A(16x4 f32) * B(4x16 f32) + C(16x16 f32)
  return __builtin_amdgcn_wmma_f32_16x16x4_f32(
      /*neg_a=*/false, a, /*neg_b=*/false, b,
      /*c_mod=*/(short)0, c, /*reuse_a=*/false, /*reuse_b=*/false);
}

// ---------------------------------------------------------------------------
// Stage 1: fused 3-layer node MLP via f32 WMMA.
//   x1 = relu(jets * Wu^T + bu)   [16384,16] x [16,64]
//   x2 = relu(x1   * Wu1^T + bu1) [16384,64] x [64,64]
//   y  =       x2  * We^T         [16384,64] x [64,64]
// Each wave owns a 16-row M tile; activations staged through padded LDS.
// ---------------------------------------------------------------------------
__global__ __launch_bounds__(256) void fused_mlp_wmma(
    const float* __restrict__ jets,
    const float* __restrict__ Wu,  const float* __restrict__ bu,
    const float* __restrict__ Wu1, const float* __restrict__ bu1,
    const float* __restrict__ We,
    float* __restrict__ y) {
  __shared__ float xs[8][16][65];  // 65-float row stride -> conflict-free col reads

  const int lane = threadIdx.x & 31;
  const int wave = threadIdx.x >> 5;
  const int ml   = lane & 15;   // A: row M;  B/D: column N (within 16-tile)
  const int hi   = lane >> 4;   // half select (0: K/M+0, 1: K+2 / M+8)
  const int koff = hi * 2;
  const int m0   = (blockIdx.x * 8 + wave) * 16;  // global row base of tile

  // ---- GEMM1: K=16 (4 WMMA k-steps), N=64 (4 n-tiles) -----------------
  v2f a1[4];
#pragma unroll
  for (int kt = 0; kt < 4; ++kt) {
    const float* row = jets + (size_t)(m0 + ml) * 16 + kt * 4 + koff;
    a1[kt].x = row[0];
    a1[kt].y = row[1];
  }
#pragma unroll
  for (int nt = 0; nt < 4; ++nt) {
    const int h0 = nt * 16;
    v8f acc = {};
#pragma unroll
    for (int kt = 0; kt < 4; ++kt) {
      const float* wrow = Wu + (size_t)(h0 + ml) * 16 + kt * 4 + koff;
      v2f bf;
      bf.x = wrow[0];
      bf.y = wrow[1];
      acc = wmma_f32_k4(a1[kt], bf, acc);
    }
    const float bv = bu[h0 + ml];
#pragma unroll
    for (int r = 0; r < 8; ++r) {
      xs[wave][r + hi * 8][h0 + ml] = fmaxf(acc[r] + bv, 0.0f);
    }
  }
  __syncthreads();

  // ---- GEMM2: K=64 (16 k-steps), read x1 from LDS ---------------------
  v8f acc2[4];
#pragma unroll
  for (int nt = 0; nt < 4; ++nt) {
    const int h0 = nt * 16;
    v8f acc = {};
#pragma unroll
    for (int kt = 0; kt < 16; ++kt) {
      const int k = kt * 4 + koff;
      v2f af;
      af.x = xs[wave][ml][k];
      af.y = xs[wave][ml][k + 1];
      const float* wrow = Wu1 + (size_t)(h0 + ml) * 64 + k;
      v2f bf;
      bf.x = wrow[0];
      bf.y = wrow[1];
      acc = wmma_f32_k4(af, bf, acc);
    }
    acc2[nt] = acc;
  }
  __syncthreads();  // all reads of x1 done before overwriting with x2
#pragma unroll
  for (int nt = 0; nt < 4; ++nt) {
    const int h0 = nt * 16;
    const float bv = bu1[h0 + ml];
#pragma unroll
    for (int r = 0; r < 8; ++r) {
      xs[wave][r + hi * 8][h0 + ml] = fmaxf(acc2[nt][r] + bv, 0.0f);
    }
  }
  __syncthreads();

  // ---- GEMM3: y = x2 * We^T (no bias here; b_edge folded into stage 2)
#pragma unroll
  for (int nt = 0; nt < 4; ++nt) {
    const int h0 = nt * 16;
    v8f acc = {};
#pragma unroll
    for (int kt = 0; kt < 16; ++kt) {
      const int k = kt * 4 + koff;
      v2f af;
      af.x = xs[wave][ml][k];
      af.y = xs[wave][ml][k + 1];
      const float* wrow = We + (size_t)(h0 + ml) * 64 + k;
      v2f bf;
      bf.x = wrow[0];
      bf.y = wrow[1];
      acc = wmma_f32_k4(af, bf, acc);
    }
#pragma unroll
    for (int r = 0; r < 8; ++r) {
      y[(size_t)(m0 + r + hi * 8) * 64 + (h0 + ml)] = acc[r];
    }
  }
}

// ---------------------------------------------------------------------------
// Stage 2: pairwise tanh + pooled partial sums, exploiting symmetry:
//   tanh(y_i + y_j + c) == tanh(y_j + y_i + c)
// so ordered-pair sum = sum(diag tiles, w=1) + 2 * sum(upper tiles, w=2).
// 32x32 (i,j) tiles over 8x8 tile grid -> 36 of 64 tile-pairs computed.
// Block = (b, h-half, tile-pair); two 32x33 LDS tiles, conflict-free reads.
// ---------------------------------------------------------------------------
__global__ __launch_bounds__(256) void pair_tanh_sym(
    const float* __restrict__ y, const float* __restrict__ b_edge,
    float* __restrict__ partial) {
  __shared__ float ysi[32][33];
  __shared__ float ysj[32][33];
  __shared__ float red[256];

  const int b  = blockIdx.x;  // 0..63
  const int hh = blockIdx.y;  // 0..1
  const int tp = blockIdx.z;  // 0..35  (upper-triangle tile pair, row-major)
  int rem = tp;
  int ti = 0;
  while (rem >= 8 - ti) {
    rem -= (8 - ti);
    ++ti;
  }
  const int tj = ti + rem;  // ti <= tj

  const int tid = threadIdx.x;
  const int h   = tid & 31;
  const int ig  = tid >> 5;  // 0..7

  // Load both 32x32 tiles (one float4 per thread per tile).
  {
    const int row = tid >> 3;       // 0..31
    const int c4  = (tid & 7) * 4;  // 0,4,...,28
    const float* srci =
        y + ((size_t)b * 256 + ti * 32 + row) * 64 + hh * 32 + c4;
    const float* srcj =
        y + ((size_t)b * 256 + tj * 32 + row) * 64 + hh * 32 + c4;
    *(float4*)(&ysi[row][c4]) = *(const float4*)srci;
    *(float4*)(&ysj[row][c4]) = *(const float4*)srcj;
  }
  __syncthreads();

  const float c = b_edge[hh * 32 + h];
  float acc0 = 0.0f, acc1 = 0.0f;  // two chains: keep TRANS pipe fed
#pragma unroll
  for (int ii = 0; ii < 4; ++ii) {
    const int i = ig * 4 + ii;
    const float yic = ysi[i][h] + c;
#pragma unroll
    for (int j = 0; j < 32; j += 2) {
      acc0 += fast_tanh(yic + ysj[j][h]);
      acc1 += fast_tanh(yic + ysj[j + 1][h]);
    }
  }
  const float w = (ti == tj) ? 1.0f : 2.0f;  // off-diag tiles count twice
  red[tid] = (acc0 + acc1) * w;
  __syncthreads();

  if (tid < 32) {
    float s = 0.0f;
#pragma unroll
    for (int g = 0; g < 8; ++g) s += red[g * 32 + h];
    partial[(((size_t)b * 2 + hh) * 36 + tp) * 32 + h] = s;
  }
}

// ---------------------------------------------------------------------------
// Stage 3: deterministic final reduction + mean scale (1 / N^2).
// ---------------------------------------------------------------------------
__global__ __launch_bounds__(64) void reduce_out(
    const float* __restrict__ partial, float* __restrict__ out) {
  const int b   = blockIdx.x;
  const int h   = threadIdx.x;  // 0..63
  const int hh  = h >> 5;
  const int h32 = h & 31;
  float s = 0.0f;
#pragma unroll
  for (int t = 0; t < 36; ++t) {
    s += partial[(((size_t)b * 2 + hh) * 36 + t) * 32 + h32];
  }
  out[(size_t)b * 64 + h] = s * (1.0f / 65536.0f);
}

// ---------------------------------------------------------------------------
extern "C" void kernel_launch(void* const* d_in, const int* in_sizes, int n_in,
                              void* d_out, int out_size, void* d_ws, size_t ws_size,
                              hipStream_t stream) {
  const float* jets = (const float*)d_in[0];  // [64,256,16]
  const float* Wu   = (const float*)d_in[1];  // [64,16]
  const float* bu   = (const float*)d_in[2];  // [64]
  const float* Wu1  = (const float*)d_in[3];  // [64,64]
  const float* bu1  = (const float*)d_in[4];  // [64]
  const float* We   = (const float*)d_in[5];  // [64,64]
  const float* be   = (const float*)d_in[6];  // [64]

  float* y       = (float*)d_ws;            // 16384*64 f32 = 4 MB
  float* partial = y + (size_t)16384 * 64;  // 64*2*36*32 f32 = 576 KB
  float* out     = (float*)d_out;           // [64,64] f32

  // Stage 1: 16384 rows / (16 rows * 8 waves) = 128 blocks.
  fused_mlp_wmma<<<128, 256, 0, stream>>>(jets, Wu, bu, Wu1, bu1, We, y);
  // Stage 2: 64 batches x 2 h-halves x 36 tile-pairs = 4608 blocks.
  pair_tanh_sym<<<dim3(64, 2, 36), 256, 0, stream>>>(y, be, partial);
  // Stage 3: final reduce.
  reduce_out<<<64, 64, 0, stream>>>(partial, out);
}